// BinaryClassifierChain_81939386073106
// MI455X (gfx1250) — compile-verified
//
#include <hip/hip_runtime.h>
#include <hip/hip_bf16.h>
#include <math.h>

typedef float v2f __attribute__((ext_vector_type(2)));
typedef float v8f __attribute__((ext_vector_type(8)));
typedef unsigned int u32x4 __attribute__((ext_vector_type(4)));
typedef int i32x4 __attribute__((ext_vector_type(4)));
typedef int i32x8 __attribute__((ext_vector_type(8)));

#define EMBED_DIM 1024
#define NUM_WORD_CLASSES 64
#define NUM_BIN 32
#define BSZ 64
#define NUM_WORDS 1024
#define KBASE (EMBED_DIM + NUM_WORD_CLASSES)   // 1088
#define D_IN (KBASE + NUM_BIN)                 // 1120
#define MROWS (BSZ * NUM_WORDS)                // 65536
#define NPAIR (NUM_WORDS * NUM_WORD_CLASSES)   // 65536

#define KC 64
#define NCHUNK (KBASE / KC)    // 17
#define LDSS 66                // 64 + 2 DWORD pad (matches TDM pad feature)
#define A_BUF_FLOATS (128 * LDSS)  // 8448 floats = 33792 B
#define W_BUF_FLOATS (32 * LDSS)   // 2112 floats =  8448 B
#define GEMM_LDS_BYTES ((2 * A_BUF_FLOATS + 2 * W_BUF_FLOATS) * 4)  // 84480 B

// ---------------------------------------------------------------------------
// Kernel 1: softmax over the batch axis (axis=0). One thread per (word,class)
// pair, 64 batch values held in VGPRs, coalesced loads/stores per b-slice.
// ---------------------------------------------------------------------------
__global__ __launch_bounds__(256) void bcc_softmax_batch(
    const float* __restrict__ wcf, float* __restrict__ wc) {
  int pair = blockIdx.x * blockDim.x + threadIdx.x;  // 0..65535
  float v[BSZ];
#pragma unroll
  for (int b = 0; b < BSZ; ++b) v[b] = wcf[(size_t)b * NPAIR + pair];
  float mx = v[0];
#pragma unroll
  for (int b = 1; b < BSZ; ++b) mx = fmaxf(mx, v[b]);
  float s = 0.0f;
#pragma unroll
  for (int b = 0; b < BSZ; ++b) { v[b] = __expf(v[b] - mx); s += v[b]; }
  float inv = 1.0f / s;
#pragma unroll
  for (int b = 0; b < BSZ; ++b) wc[(size_t)b * NPAIR + pair] = v[b] * inv;
}

// ---------------------------------------------------------------------------
// TDM helper: issue TENSOR_LOAD_TO_LDS of a 2D tile (tile_d0 x tile_d1 f32
// elements, row stride stride_d0 elements) with LDS row padding of 2 DWORDs
// per 64 DWORDs (-> 66-float LDS row stride). D# bitfields per CDNA5 ISA
// §8.3/§8.4. This toolchain uses the clang-23 6-arg builtin form.
// ---------------------------------------------------------------------------
__device__ __forceinline__ unsigned lds_off(const void* p) {
  // LDS aperture flat pointers carry the LDS byte offset in the low 32 bits.
  return (unsigned)(unsigned long long)(uintptr_t)p;
}

__device__ __forceinline__ void tdm_load_2d_f32(
    const void* gptr, unsigned lds_byte,
    unsigned tensor_d0, unsigned tensor_d1,
    unsigned tile_d0, unsigned tile_d1, unsigned stride_d0) {
  unsigned long long ga = (unsigned long long)(uintptr_t)gptr;
  u32x4 g0;
  g0.x = 1u;                                        // count=1, user mode
  g0.y = lds_byte;                                  // lds_addr
  g0.z = (unsigned)(ga & 0xFFFFFFFFu);              // global_addr[31:0]
  g0.w = (unsigned)((ga >> 32) & 0x01FFFFFFu)       // global_addr[56:32]
         | (2u << 30);                              // type=2 ("image")
  i32x8 g1;
  g1[0] = (2 << 16)      // data_size = 4 bytes
        | (1 << 20)      // pad_enable
        | (5 << 22)      // pad_interval: 64 DWORDs
        | (1 << 25);     // pad_amount: 2 DWORDs
  g1[1] = (int)((tensor_d0 & 0xFFFFu) << 16);                        // dim0 lo
  g1[2] = (int)((tensor_d0 >> 16) | ((tensor_d1 & 0xFFFFu) << 16));  // dim0 hi | dim1 lo
  g1[3] = (int)((tensor_d1 >> 16) | (tile_d0 << 16));                // dim1 hi | tile0
  g1[4] = (int)(tile_d1 & 0xFFFFu);                                  // tile1, tile2=0
  g1[5] = (int)stride_d0;                                            // dim0_stride lo
  g1[6] = 0;
  g1[7] = 0;
  i32x4 z4 = {0, 0, 0, 0};
  i32x8 z8 = {0, 0, 0, 0, 0, 0, 0, 0};
  __builtin_amdgcn_tensor_load_to_lds(g0, g1, z4, z4, z8, 0);
}

// ---------------------------------------------------------------------------
// Kernel 2: streaming skinny GEMM  logits = [features|wc] x W[:, :1088]^T + b
// V_WMMA_F32_16X16X4_F32, TDM-fed double-buffered LDS pipeline.
// Block = 256 threads = 8 wave32s; each wave owns a 16-row M tile and both
// 16-col N tiles. K chunked by 64; chunk c+1 streams via TDM while chunk c
// feeds the matrix units (s_wait_tensorcnt 2 = previous pair complete).
// ---------------------------------------------------------------------------
__global__ __launch_bounds__(256) void bcc_gemm_wmma(
    const float* __restrict__ feat, const float* __restrict__ wc,
    const float* __restrict__ W, const float* __restrict__ bias,
    float* __restrict__ logits) {
  extern __shared__ float smem[];

  const int tid  = threadIdx.x;
  const int lane = tid & 31;
  const int wave = tid >> 5;
  const int mBase = blockIdx.x * 128;

  // LDS byte offsets of the four buffers (runtime arithmetic only — no
  // pointer arrays, which lld cannot statically initialize).
  const unsigned ldsA0 = lds_off(smem);
  const unsigned ldsA1 = ldsA0 + A_BUF_FLOATS * 4u;
  const unsigned ldsW0 = ldsA0 + 2u * A_BUF_FLOATS * 4u;
  const unsigned ldsW1 = ldsW0 + W_BUF_FLOATS * 4u;

  v8f acc0 = {};
  v8f acc1 = {};

  const int mloc = wave * 16 + (lane & 15);  // wave-local A row
  const int half = lane >> 4;                // A: K{0,1} vs K{2,3}
  const int nn0  = lane & 15;                // B/C column within tile

  // Prime the pipeline: chunk 0 tiles in flight.
  if (wave == 0) {
    tdm_load_2d_f32(feat + (size_t)mBase * EMBED_DIM, ldsA0,
                    EMBED_DIM, MROWS, KC, 128, EMBED_DIM);
    tdm_load_2d_f32(W, ldsW0, D_IN, NUM_BIN, KC, NUM_BIN, D_IN);
  }

  for (int chunk = 0; chunk < NCHUNK; ++chunk) {
    const int bsel = chunk & 1;
    if (wave == 0) {
      const int nxt = chunk + 1;
      const unsigned ldsAn = (bsel ^ 1) ? ldsA1 : ldsA0;
      const unsigned ldsWn = (bsel ^ 1) ? ldsW1 : ldsW0;
      if (nxt < NCHUNK) {
        // Issue next chunk into the other buffer (safe: that buffer was
        // released by the barrier that ended iteration chunk-1).
        if (nxt < 16) {
          tdm_load_2d_f32(feat + (size_t)mBase * EMBED_DIM + nxt * KC,
                          ldsAn, EMBED_DIM, MROWS, KC, 128, EMBED_DIM);
        } else {
          tdm_load_2d_f32(wc + (size_t)mBase * NUM_WORD_CLASSES,
                          ldsAn, NUM_WORD_CLASSES, MROWS, KC, 128,
                          NUM_WORD_CLASSES);
        }
        tdm_load_2d_f32(W + nxt * KC, ldsWn, D_IN, NUM_BIN, KC, NUM_BIN, D_IN);
        __builtin_amdgcn_s_wait_tensorcnt(2);  // chunk `chunk` pair complete
      } else {
        __builtin_amdgcn_s_wait_tensorcnt(0);  // last pair complete
      }
    }
    __syncthreads();

    const float* sA = smem + bsel * A_BUF_FLOATS;
    const float* sW = smem + 2 * A_BUF_FLOATS + bsel * W_BUF_FLOATS;
#pragma unroll
    for (int s = 0; s < 16; ++s) {
      const int k0 = 4 * s;
      v2f a;  // A frag: lane holds A[mloc][k0+2*half .. +1]
      a.x = sA[mloc * LDSS + k0 + 2 * half];
      a.y = sA[mloc * LDSS + k0 + 2 * half + 1];
      v2f b0, b1;  // B frag: lane holds W^T[k0+half][n], W^T[k0+half+2][n]
      b0.x = sW[nn0 * LDSS + k0 + half];
      b0.y = sW[nn0 * LDSS + k0 + half + 2];
      b1.x = sW[(16 + nn0) * LDSS + k0 + half];
      b1.y = sW[(16 + nn0) * LDSS + k0 + half + 2];
      acc0 = __builtin_amdgcn_wmma_f32_16x16x4_f32(
          false, a, false, b0, (short)0, acc0, false, false);
      acc1 = __builtin_amdgcn_wmma_f32_16x16x4_f32(
          false, a, false, b1, (short)0, acc1, false, false);
    }
    __syncthreads();
  }

  // --- epilogue: bias add + store logits (M x 32) ---
  const float bias0 = bias[nn0];
  const float bias1 = bias[16 + nn0];
  const int mrowBase = mBase + wave * 16 + half * 8;  // C layout: VGPR r -> M = 8*half + r
#pragma unroll
  for (int r = 0; r < 8; ++r) {
    const size_t m = (size_t)(mrowBase + r);
    logits[m * NUM_BIN + nn0]      = acc0[r] + bias0;
    logits[m * NUM_BIN + 16 + nn0] = acc1[r] + bias1;
  }
}

// ---------------------------------------------------------------------------
// Kernel 3: serial 32-step sigmoid chain, one row per thread. Wbin in LDS.
// probs[j]==0 for j>=i makes the full j-loop equal to the triangular sum.
// ---------------------------------------------------------------------------
__global__ __launch_bounds__(256) void bcc_chain(
    const float* __restrict__ W, const float* __restrict__ logits,
    float* __restrict__ out) {
  __shared__ float sWb[NUM_BIN * 33];
  const int tid = threadIdx.x;
#pragma unroll
  for (int i = 0; i < 4; ++i) {
    int slot = tid + i * 256;            // 0..1023
    int r = slot >> 5, c = slot & 31;
    sWb[r * 33 + c] = W[(size_t)r * D_IN + KBASE + c];
  }
  __syncthreads();

  const size_t row = (size_t)blockIdx.x * blockDim.x + tid;
  float lg[NUM_BIN];
#pragma unroll
  for (int i = 0; i < 8; ++i) {
    float4 v = *(const float4*)(logits + row * NUM_BIN + i * 4);
    lg[i * 4 + 0] = v.x; lg[i * 4 + 1] = v.y;
    lg[i * 4 + 2] = v.z; lg[i * 4 + 3] = v.w;
  }
  float p[NUM_BIN];
#pragma unroll
  for (int i = 0; i < NUM_BIN; ++i) p[i] = 0.0f;
#pragma unroll
  for (int i = 0; i < NUM_BIN; ++i) {
    float z = lg[i];
#pragma unroll
    for (int j = 0; j < NUM_BIN; ++j) z = fmaf(p[j], sWb[i * 33 + j], z);
    p[i] = 1.0f / (1.0f + __expf(-z));
  }
#pragma unroll
  for (int i = 0; i < 8; ++i) {
    float4 v;
    v.x = p[i * 4 + 0]; v.y = p[i * 4 + 1];
    v.z = p[i * 4 + 2]; v.w = p[i * 4 + 3];
    *(float4*)(out + row * NUM_BIN + i * 4) = v;
  }
}

// ---------------------------------------------------------------------------
extern "C" void kernel_launch(void* const* d_in, const int* in_sizes, int n_in,
                              void* d_out, int out_size, void* d_ws, size_t ws_size,
                              hipStream_t stream) {
  const float* feat = (const float*)d_in[0];  // (64,1024,1024)
  const float* wcf  = (const float*)d_in[1];  // (64,1024,64)
  const float* W    = (const float*)d_in[2];  // (32,1120)
  const float* bias = (const float*)d_in[3];  // (32,)
  float* out = (float*)d_out;                 // (64,1024,32)

  float* wc     = (float*)d_ws;                                              // 16 MB
  float* logits = (float*)((char*)d_ws + (size_t)BSZ * NPAIR * sizeof(float)); // 8 MB

  // 84480 B dynamic LDS (> 64KB default cap; CDNA5 WGP has 320KB).
  (void)hipFuncSetAttribute((const void*)bcc_gemm_wmma,
                            hipFuncAttributeMaxDynamicSharedMemorySize,
                            GEMM_LDS_BYTES);

  bcc_softmax_batch<<<NPAIR / 256, 256, 0, stream>>>(wcf, wc);
  bcc_gemm_wmma<<<MROWS / 128, 256, GEMM_LDS_BYTES, stream>>>(feat, wc, W, bias, logits);
  bcc_chain<<<MROWS / 256, 256, 0, stream>>>(W, logits, out);
}